// LIFSpikeIN_80075370267351
// MI455X (gfx1250) — compile-verified
//
#include <hip/hip_runtime.h>
#include <cstdint>

// LIF spike recurrence, fused over T timesteps.
//   mem = mem*TAU + x[t]; spike = (mem - THRESH > 0); mem = (1-spike)*mem
// Memory-bound: 268 MB min traffic -> ~11.5 us @ 23.3 TB/s. No matmul structure
// (WMMA inapplicable), so the CDNA5-specific path is the async global->LDS copy
// engine: all T=4 timestep tiles are issued up-front (64 B/lane in flight,
// tracked by ASYNCcnt) and drained with s_wait_asynccnt 3..0 while the membrane
// state stays in registers across the unrolled T loop.

#define TPB 256
constexpr int   T_STEPS = 4;
constexpr float THRESH  = 1.0f;
constexpr float TAU     = 1.0f;

typedef __attribute__((address_space(3))) float4 f4_lds;
typedef __attribute__((ext_vector_type(4))) float vf4;   // native vector for NT store

// Async DMA: copy 16 bytes/lane from global to LDS (gfx1250, tracked by ASYNCcnt).
__device__ __forceinline__ void async_load_b128(uint32_t lds_byte_off, const float* gaddr) {
    asm volatile("global_load_async_to_lds_b128 %0, %1, off"
                 :: "v"(lds_byte_off), "v"(gaddr)
                 : "memory");
}

template <int N>
__device__ __forceinline__ void wait_asynccnt() {
    asm volatile("s_wait_asynccnt %0" :: "n"(N) : "memory");
}

__global__ __launch_bounds__(TPB) void lif_spike_kernel(const float* __restrict__ x,
                                                        float* __restrict__ out,
                                                        int n_per_t) {
    // One 16B staging slot per lane per timestep: 4 * 256 * 16B = 16 KB LDS/block.
    __shared__ float4 tile[T_STEPS][TPB];

    const int       tid    = threadIdx.x;
    const long long e0     = ((long long)blockIdx.x * TPB + tid) * 4;
    const size_t    stride = (size_t)n_per_t;   // elements per timestep

    if (e0 + 3 < (long long)n_per_t) {
        const float* gx = x + e0;

        // Issue ALL timestep loads immediately: max MLP, zero VGPR pressure,
        // in-order completion lets each wait retire exactly one buffer.
#pragma unroll
        for (int t = 0; t < T_STEPS; ++t) {
            const uint32_t lds_off = (uint32_t)(uintptr_t)(f4_lds*)&tile[t][tid];
            async_load_b128(lds_off, gx + (size_t)t * stride);
        }

        float4 mem = make_float4(0.f, 0.f, 0.f, 0.f);

#pragma unroll
        for (int t = 0; t < T_STEPS; ++t) {
            // asynccnt <= (T-1-t)  =>  load t has landed in LDS.
            if      (t == 0) wait_asynccnt<3>();
            else if (t == 1) wait_asynccnt<2>();
            else if (t == 2) wait_asynccnt<1>();
            else             wait_asynccnt<0>();

            const float4 xt = tile[t][tid];   // ds_load_b128, lane-private slot

            // LIF update per component.
            mem.x = mem.x * TAU + xt.x;
            mem.y = mem.y * TAU + xt.y;
            mem.z = mem.z * TAU + xt.z;
            mem.w = mem.w * TAU + xt.w;

            vf4 s;
            s.x = (mem.x - THRESH > 0.f) ? 1.f : 0.f;
            s.y = (mem.y - THRESH > 0.f) ? 1.f : 0.f;
            s.z = (mem.z - THRESH > 0.f) ? 1.f : 0.f;
            s.w = (mem.w - THRESH > 0.f) ? 1.f : 0.f;

            mem.x = (1.f - s.x) * mem.x;
            mem.y = (1.f - s.y) * mem.y;
            mem.z = (1.f - s.z) * mem.z;
            mem.w = (1.f - s.w) * mem.w;

            // Spikes are write-once / never re-read here: non-temporal B128 store.
            __builtin_nontemporal_store(s, (vf4*)(out + (size_t)t * stride + e0));
        }
    } else if (e0 < (long long)n_per_t) {
        // Ragged tail (not hit for the reference shape): scalar, guarded, no async.
        for (int j = 0; j < 4; ++j) {
            const long long e = e0 + j;
            if (e >= (long long)n_per_t) break;
            float mm = 0.f;
            for (int t = 0; t < T_STEPS; ++t) {
                mm = mm * TAU + x[(size_t)t * stride + (size_t)e];
                const float s = (mm - THRESH > 0.f) ? 1.f : 0.f;
                out[(size_t)t * stride + (size_t)e] = s;
                mm = (1.f - s) * mm;
            }
        }
    }
}

extern "C" void kernel_launch(void* const* d_in, const int* in_sizes, int n_in,
                              void* d_out, int out_size, void* d_ws, size_t ws_size,
                              hipStream_t stream) {
    (void)n_in; (void)out_size; (void)d_ws; (void)ws_size;

    const float* x   = (const float*)d_in[0];
    float*       out = (float*)d_out;

    const int total   = in_sizes[0];          // T * B * C * H * W
    const int n_per_t = total / T_STEPS;      // elements per timestep
    const int nvec    = (n_per_t + 3) / 4;    // float4 units per timestep
    const int blocks  = (nvec + TPB - 1) / TPB;

    hipLaunchKernelGGL(lif_spike_kernel, dim3(blocks), dim3(TPB), 0, stream,
                       x, out, n_per_t);
}